// RegionGraphReranker_64355789963635
// MI455X (gfx1250) — compile-verified
//
#include <hip/hip_runtime.h>
#include <hip/hip_bf16.h>
#include <math.h>

// ---------------- problem constants ----------------
#define KP    1024            // pages
#define RNUM  4
#define KR    (KP * RNUM)     // regions = 4096
#define NN    (KP + KR + 1)   // nodes = 5121
#define MP    5136            // padded rows (321 * 16)
#define MT    (MP / 16)       // 321 row tiles
#define DIN   128
#define DHID  256
#define DOUT  64
#define NHEAD 4
#define SEMTH 0.7f

typedef float v2f __attribute__((ext_vector_type(2)));
typedef float v8f __attribute__((ext_vector_type(8)));

static __device__ __forceinline__ v8f wmma_f32_16x16x4(v2f a, v2f b, v8f c) {
  return __builtin_amdgcn_wmma_f32_16x16x4_f32(false, a, false, b, (short)0, c,
                                               false, false);
}

// adjacency connectivity mask (adj > 0); i, j must be < NN
static __device__ __forceinline__ bool adj_mask(int i, int j,
                                                const float* __restrict__ ps,
                                                const int* __restrict__ pn) {
  if (i == NN - 1 || j == NN - 1) return true;      // query row/col + its diag
  const bool ir = (i >= KP), jr = (j >= KP);
  if (!ir && !jr) {
    if (i == j) return true;                         // self loop
    int d = pn[i] - pn[j];
    if (d == 1 || d == -1) return true;              // page-number adjacency
    return ps[(size_t)i * KP + j] >= SEMTH;          // semantic edge
  }
  if (!ir) return ((j - KP) >> 2) == i;              // page i <-> its regions
  if (!jr) return ((i - KP) >> 2) == j;
  return ((i - KP) >> 2) == ((j - KP) >> 2);         // same-page regions (+diag)
}

// ---------------- node feature normalization ----------------
__global__ void normalize_nodes(const float* __restrict__ pv,
                                const float* __restrict__ rv,
                                const float* __restrict__ qv,
                                float* __restrict__ node) {
  const int r = blockIdx.x;        // 0..MP-1
  const int t = threadIdx.x;       // 0..127
  const float* src = nullptr;
  if (r < KP)            src = pv + (size_t)r * DIN;
  else if (r < KP + KR)  src = rv + (size_t)(r - KP) * DIN;
  else if (r == NN - 1)  src = qv;
  float v = src ? src[t] : 0.0f;
  __shared__ float red[128];
  red[t] = v * v;
  __syncthreads();
  for (int s = 64; s > 0; s >>= 1) {
    if (t < s) red[t] += red[t + s];
    __syncthreads();
  }
  float scale = 1.0f / fmaxf(sqrtf(red[0]), 1e-12f);
  node[(size_t)r * DIN + t] = v * scale;
}

// ---------------- WMMA f32 GEMM: C(M x N) = A(M x K) * B + bias ----------------
// block = 128 threads (4 waves); each wave computes a 16 x (16*NTW) strip.
// grid: (N / (64*NTW), M/16). Branch-free inner loop; loop-carried *integer*
// offsets into B (keeps global addrspace inference -> global_load_b64, not
// flat_load) so all NTW fragment loads issue as one clause before the WMMAs.
template <bool TRANSB, bool BIAS, int NTW>
__global__ void gemm16(const float* __restrict__ A, int lda,
                       const float* __restrict__ B, int ldb,
                       float* __restrict__ C, int ldc,
                       int Kd, const float* __restrict__ bias) {
  const int lane = threadIdx.x & 31;
  const int wave = threadIdx.x >> 5;
  const int lo = lane & 15, hi = lane >> 4;
  const int row0 = blockIdx.y * 16;
  const int n0 = (blockIdx.x * 4 + wave) * (16 * NTW);
  v8f acc[NTW];
#pragma unroll
  for (int t = 0; t < NTW; ++t)
    acc[t] = (v8f){0.f, 0.f, 0.f, 0.f, 0.f, 0.f, 0.f, 0.f};

  const float* ap = A + (size_t)(row0 + lo) * lda + 2 * hi;   // streamed rows
  size_t boff[NTW];
#pragma unroll
  for (int t = 0; t < NTW; ++t)
    boff[t] = TRANSB ? ((size_t)(n0 + t * 16 + lo) * ldb + 2 * hi)
                     : ((size_t)(2 * hi) * ldb + n0 + t * 16 + lo);
  const int bstride = TRANSB ? 1 : ldb;     // distance between the 2 B elems
  const int bstep = TRANSB ? 4 : 4 * ldb;   // per-k-step advance

#pragma unroll 2
  for (int kc = 0; kc < Kd; kc += 4) {
    __builtin_prefetch(ap + kc + 64, 0, 3);   // keep A stream warm near-cache
    v2f a, b[NTW];
    a.x = ap[kc];
    a.y = ap[kc + 1];
#pragma unroll
    for (int t = 0; t < NTW; ++t) {           // load phase (clauses together)
      b[t].x = B[boff[t]];
      b[t].y = B[boff[t] + bstride];
    }
#pragma unroll
    for (int t = 0; t < NTW; ++t)             // compute phase
      acc[t] = wmma_f32_16x16x4(a, b[t], acc[t]);
#pragma unroll
    for (int t = 0; t < NTW; ++t) boff[t] += bstep;
  }
#pragma unroll
  for (int t = 0; t < NTW; ++t) {
    float bv = BIAS ? bias[n0 + t * 16 + lo] : 0.0f;
#pragma unroll
    for (int r = 0; r < 8; ++r)
      C[(size_t)(row0 + r + 8 * hi) * ldc + n0 + t * 16 + lo] = acc[t][r] + bv;
  }
}

// ---------------- per-row/head attention coefficients s,d ----------------
// grid (MP, NHEAD), block 64
__global__ void compute_sd(const float* __restrict__ Wh, int F,
                           const float* __restrict__ asrc,
                           const float* __restrict__ adst,
                           float* __restrict__ sA, float* __restrict__ dA) {
  const int i = blockIdx.x, h = blockIdx.y, t = threadIdx.x;
  float ss = 0.f, dd = 0.f;
  for (int f = t; f < F; f += 64) {
    float v = Wh[(size_t)i * (NHEAD * F) + h * F + f];
    ss += v * asrc[h * F + f];
    dd += v * adst[h * F + f];
  }
  __shared__ float rs[64], rd[64];
  rs[t] = ss; rd[t] = dd;
  __syncthreads();
  for (int s = 32; s > 0; s >>= 1) {
    if (t < s) { rs[t] += rs[t + s]; rd[t] += rd[t + s]; }
    __syncthreads();
  }
  if (t == 0) { sA[i * NHEAD + h] = rs[0]; dA[i * NHEAD + h] = rd[0]; }
}

// ---------------- softmax row stats (max m, denom l) ----------------
// grid (MP), block 256
__global__ void attn_stats(const float* __restrict__ sA,
                           const float* __restrict__ dA,
                           const float* __restrict__ ps,
                           const int* __restrict__ pn,
                           float* __restrict__ mA, float* __restrict__ lA) {
  const int i = blockIdx.x;
  const int t = threadIdx.x;
  if (i >= NN) {                      // padded rows: benign stats
    if (t < NHEAD) { mA[i * NHEAD + t] = 0.f; lA[i * NHEAD + t] = 1.f; }
    return;
  }
  float si[NHEAD];
#pragma unroll
  for (int h = 0; h < NHEAD; ++h) si[h] = sA[i * NHEAD + h];
  float mm[NHEAD], ll[NHEAD];
#pragma unroll
  for (int h = 0; h < NHEAD; ++h) { mm[h] = -1e30f; ll[h] = 0.f; }
  for (int j = t; j < NN; j += 256) {
    if (!adj_mask(i, j, ps, pn)) continue;
#pragma unroll
    for (int h = 0; h < NHEAD; ++h) {
      float e = si[h] + dA[j * NHEAD + h];
      e = (e >= 0.f) ? e : 0.2f * e;                 // leaky relu
      if (e > mm[h]) {
        ll[h] = ll[h] * __expf(mm[h] - e) + 1.f;
        mm[h] = e;
      } else {
        ll[h] += __expf(e - mm[h]);
      }
    }
  }
  __shared__ float shm[256], shl[256];
  for (int h = 0; h < NHEAD; ++h) {
    shm[t] = mm[h]; shl[t] = ll[h];
    __syncthreads();
    for (int s = 128; s > 0; s >>= 1) {
      if (t < s) {
        float m1 = shm[t], m2 = shm[t + s];
        float l1 = shl[t], l2 = shl[t + s];
        float mn = fmaxf(m1, m2);
        shm[t] = mn;
        shl[t] = l1 * __expf(m1 - mn) + l2 * __expf(m2 - mn);
      }
      __syncthreads();
    }
    if (t == 0) { mA[i * NHEAD + h] = shm[0]; lA[i * NHEAD + h] = shl[0]; }
    __syncthreads();
  }
}

// ---------------- attention aggregation: Out[i,h,:] = sum_j att * Wh[j,h,:] ----
// grid (NHEAD, MT), block 32 (single wave). FPH = features per head.
template <int FPH>
__global__ void attn_agg(const float* __restrict__ Wh,
                         const float* __restrict__ sA,
                         const float* __restrict__ dA,
                         const float* __restrict__ mA,
                         const float* __restrict__ lA,
                         const float* __restrict__ ps,
                         const int* __restrict__ pn,
                         float* __restrict__ Out) {
  constexpr int NT = FPH / 16;
  const int h = blockIdx.x;
  const int row0 = blockIdx.y * 16;
  const int lane = threadIdx.x;
  const int lo = lane & 15, hi = lane >> 4;
  const int ldWh = NHEAD * FPH;
  __shared__ float P[256];                       // 16x16 att tile, row-major i x j
  v8f acc[NT];
#pragma unroll
  for (int t = 0; t < NT; ++t) acc[t] = (v8f){0.f,0.f,0.f,0.f,0.f,0.f,0.f,0.f};
  // loop-carried integer offset of Wh fragment base (row 2*hi, head column blk)
  size_t woff = (size_t)(2 * hi) * ldWh + h * FPH + lo;
  for (int jt = 0; jt < MT; ++jt) {
    const int j0 = jt * 16;
#pragma unroll
    for (int t = 0; t < 8; ++t) {
      int idx = lane + 32 * t;
      int gi = row0 + (idx >> 4), gj = j0 + (idx & 15);
      float att = 0.f;
      if (gi < NN && gj < NN && adj_mask(gi, gj, ps, pn)) {
        float e = sA[gi * NHEAD + h] + dA[gj * NHEAD + h];
        e = (e >= 0.f) ? e : 0.2f * e;
        att = __expf(e - mA[gi * NHEAD + h]) / lA[gi * NHEAD + h];
      }
      P[idx] = att;
    }
    __syncthreads();
#pragma unroll
    for (int w = 0; w < 4; ++w) {
      v2f a;
      a.x = P[lo * 16 + 4 * w + 2 * hi];
      a.y = P[lo * 16 + 4 * w + 2 * hi + 1];
      const size_t bo = woff + (size_t)(4 * w) * ldWh;
      v2f b[NT];
#pragma unroll
      for (int ft = 0; ft < NT; ++ft) {        // load phase
        b[ft].x = Wh[bo + ft * 16];
        b[ft].y = Wh[bo + ft * 16 + ldWh];
      }
#pragma unroll
      for (int ft = 0; ft < NT; ++ft)          // compute phase
        acc[ft] = wmma_f32_16x16x4(a, b[ft], acc[ft]);
    }
    __syncthreads();
    woff += (size_t)16 * ldWh;
  }
#pragma unroll
  for (int ft = 0; ft < NT; ++ft)
#pragma unroll
    for (int r = 0; r < 8; ++r)
      Out[(size_t)(row0 + r + 8 * hi) * ldWh + h * FPH + ft * 16 + lo] =
          acc[ft][r];
}

// ---------------- elementwise kernels ----------------
__global__ void elu_k(float* __restrict__ x, size_t n) {
  size_t i = (size_t)blockIdx.x * blockDim.x + threadIdx.x;
  if (i < n) {
    float v = x[i];
    x[i] = (v > 0.f) ? v : (__expf(v) - 1.f);
  }
}

__global__ void combine_k(const float* __restrict__ out1,
                          const float* __restrict__ hres,
                          float* __restrict__ h2) {
  size_t idx = (size_t)blockIdx.x * blockDim.x + threadIdx.x;
  if (idx >= (size_t)MP * DOUT) return;
  size_t i = idx / DOUT, f = idx % DOUT;
  const float* o = out1 + i * (NHEAD * DOUT);
  float m = 0.25f * (o[f] + o[DOUT + f] + o[2 * DOUT + f] + o[3 * DOUT + f]);
  h2[idx] = m + hres[idx];
}

// ---------------- scoring MLP ----------------
// grid (KP), block 64
__global__ void score_k(const float* __restrict__ h2,
                        const float* __restrict__ Ws1,
                        const float* __restrict__ bs1,
                        const float* __restrict__ Ws2,
                        const float* __restrict__ bs2,
                        float* __restrict__ delta) {
  const int i = blockIdx.x, f = threadIdx.x;
  const float* qr = h2 + (size_t)(NN - 1) * DOUT;
  const float* pr = h2 + (size_t)i * DOUT;
  float acc = bs1[f];
  for (int k = 0; k < DOUT; ++k) acc += pr[k] * Ws1[k * DOUT + f];
  for (int k = 0; k < DOUT; ++k) acc += qr[k] * Ws1[(DOUT + k) * DOUT + f];
  acc = fmaxf(acc, 0.f);
  float v = acc * Ws2[f];
  __shared__ float red[64];
  red[f] = v;
  __syncthreads();
  for (int s = 32; s > 0; s >>= 1) {
    if (f < s) red[f] += red[f + s];
    __syncthreads();
  }
  if (f == 0) delta[i] = red[0] + bs2[0];
}

__global__ void minmax_k(const float* __restrict__ s0, float* __restrict__ mm) {
  const int t = threadIdx.x;  // 256
  float mn = 1e30f, mx = -1e30f;
  for (int i = t; i < KP; i += 256) {
    float v = s0[i];
    mn = fminf(mn, v);
    mx = fmaxf(mx, v);
  }
  __shared__ float rn[256], rx[256];
  rn[t] = mn; rx[t] = mx;
  __syncthreads();
  for (int s = 128; s > 0; s >>= 1) {
    if (t < s) {
      rn[t] = fminf(rn[t], rn[t + s]);
      rx[t] = fmaxf(rx[t], rx[t + s]);
    }
    __syncthreads();
  }
  if (t == 0) { mm[0] = rn[0]; mm[1] = rx[0]; }
}

__global__ void final_k(const float* __restrict__ s0,
                        const float* __restrict__ delta,
                        const float* __restrict__ mm,
                        const float* __restrict__ lam_p,
                        float* __restrict__ out) {
  int i = blockIdx.x * blockDim.x + threadIdx.x;
  if (i >= KP) return;
  float lam = 1.f / (1.f + __expf(-lam_p[0]));
  float s = (s0[i] - mm[0]) / (mm[1] - mm[0] + 1e-8f);
  out[i] = (1.f - lam) * s + lam * delta[i];
}

// ---------------- host launch ----------------
extern "C" void kernel_launch(void* const* d_in, const int* in_sizes, int n_in,
                              void* d_out, int out_size, void* d_ws,
                              size_t ws_size, hipStream_t stream) {
  const float* pv    = (const float*)d_in[0];   // (K, 128)
  const float* rv    = (const float*)d_in[1];   // (K*R, 128)
  const float* qv    = (const float*)d_in[2];   // (128,)
  const int*   pn    = (const int*)  d_in[3];   // (K,)
  const float* s1sc  = (const float*)d_in[4];   // (K,)
  const float* W_in  = (const float*)d_in[5];   // (128, 256)
  const float* b_in  = (const float*)d_in[6];   // (256,)
  const float* W_g0  = (const float*)d_in[7];   // (256, 1024)
  const float* as0   = (const float*)d_in[8];   // (4, 256)
  const float* ad0   = (const float*)d_in[9];   // (4, 256)
  const float* W_g1  = (const float*)d_in[10];  // (1024, 256)
  const float* as1   = (const float*)d_in[11];  // (4, 64)
  const float* ad1   = (const float*)d_in[12];  // (4, 64)
  const float* W_res = (const float*)d_in[13];  // (256, 64)
  const float* b_res = (const float*)d_in[14];  // (64,)
  const float* W_s1  = (const float*)d_in[15];  // (128, 64)
  const float* b_s1  = (const float*)d_in[16];  // (64,)
  const float* W_s2  = (const float*)d_in[17];  // (64, 1)
  const float* b_s2  = (const float*)d_in[18];  // (1,)
  const float* lam   = (const float*)d_in[19];  // scalar
  float* out = (float*)d_out;

  // workspace layout (floats)
  float* wsf = (float*)d_ws;
  size_t o = 0;
  float* node = wsf + o; o += (size_t)MP * DIN;            // 5136x128
  float* psb  = wsf + o; o += (size_t)KP * KP;             // 1024x1024
  float* hbuf = wsf + o; o += (size_t)MP * DHID;           // 5136x256
  float* Wh0  = wsf + o; o += (size_t)MP * (NHEAD * DHID); // 5136x1024
  float* out0 = wsf + o; o += (size_t)MP * (NHEAD * DHID); // 5136x1024 (-> h1)
  float* Wh1  = wsf + o; o += (size_t)MP * (NHEAD * DOUT); // 5136x256
  float* out1 = wsf + o; o += (size_t)MP * (NHEAD * DOUT); // 5136x256
  float* hres = wsf + o; o += (size_t)MP * DOUT;           // 5136x64
  float* h2   = wsf + o; o += (size_t)MP * DOUT;           // 5136x64
  float* s0a  = wsf + o; o += (size_t)MP * NHEAD;
  float* d0a  = wsf + o; o += (size_t)MP * NHEAD;
  float* m0   = wsf + o; o += (size_t)MP * NHEAD;
  float* l0   = wsf + o; o += (size_t)MP * NHEAD;
  float* s1a  = wsf + o; o += (size_t)MP * NHEAD;
  float* d1a  = wsf + o; o += (size_t)MP * NHEAD;
  float* m1   = wsf + o; o += (size_t)MP * NHEAD;
  float* l1   = wsf + o; o += (size_t)MP * NHEAD;
  float* dlt  = wsf + o; o += (size_t)KP;
  float* mmv  = wsf + o; o += 2;

  // 1. normalized node features (padded rows zero)
  normalize_nodes<<<dim3(MP), dim3(128), 0, stream>>>(pv, rv, qv, node);

  // 2. ps = pvn @ pvn^T  (1024x1024, K=128)   [WMMA, B transposed]
  gemm16<true, false, 4><<<dim3(KP / 256, KP / 16), dim3(128), 0, stream>>>(
      node, DIN, node, DIN, psb, KP, DIN, nullptr);

  // 3. h = node @ W_in + b_in  (5136x256, K=128)   [WMMA]
  gemm16<false, true, 4><<<dim3(1, MT), dim3(128), 0, stream>>>(
      node, DIN, W_in, DHID, hbuf, DHID, DIN, b_in);

  // 4. Wh0 = h @ W_g0  (5136x1024, K=256)   [WMMA]
  gemm16<false, false, 4><<<dim3(4, MT), dim3(128), 0, stream>>>(
      hbuf, DHID, W_g0, NHEAD * DHID, Wh0, NHEAD * DHID, DHID, nullptr);

  // 5-7. GAT layer 0 attention
  compute_sd<<<dim3(MP, NHEAD), dim3(64), 0, stream>>>(Wh0, DHID, as0, ad0,
                                                       s0a, d0a);
  attn_stats<<<dim3(MP), dim3(256), 0, stream>>>(s0a, d0a, psb, pn, m0, l0);
  attn_agg<DHID><<<dim3(NHEAD, MT), dim3(32), 0, stream>>>(
      Wh0, s0a, d0a, m0, l0, psb, pn, out0);

  // 8. h1 = elu(out0) in place
  {
    size_t n = (size_t)MP * (NHEAD * DHID);
    elu_k<<<dim3((unsigned)((n + 255) / 256)), dim3(256), 0, stream>>>(out0, n);
  }

  // 9. Wh1 = h1 @ W_g1  (5136x256, K=1024)   [WMMA]
  gemm16<false, false, 4><<<dim3(1, MT), dim3(128), 0, stream>>>(
      out0, NHEAD * DHID, W_g1, NHEAD * DOUT, Wh1, NHEAD * DOUT, NHEAD * DHID,
      nullptr);

  // 10-12. GAT layer 1 attention
  compute_sd<<<dim3(MP, NHEAD), dim3(64), 0, stream>>>(Wh1, DOUT, as1, ad1,
                                                       s1a, d1a);
  attn_stats<<<dim3(MP), dim3(256), 0, stream>>>(s1a, d1a, psb, pn, m1, l1);
  attn_agg<DOUT><<<dim3(NHEAD, MT), dim3(32), 0, stream>>>(
      Wh1, s1a, d1a, m1, l1, psb, pn, out1);

  // 13. residual: hres = h @ W_res + b_res  (5136x64, K=256)   [WMMA]
  gemm16<false, true, 1><<<dim3(1, MT), dim3(128), 0, stream>>>(
      hbuf, DHID, W_res, DOUT, hres, DOUT, DHID, b_res);

  // 14. h2 = mean_heads(out1) + hres
  {
    size_t n = (size_t)MP * DOUT;
    combine_k<<<dim3((unsigned)((n + 255) / 256)), dim3(256), 0, stream>>>(
        out1, hres, h2);
  }

  // 15. scoring MLP -> delta
  score_k<<<dim3(KP), dim3(64), 0, stream>>>(h2, W_s1, b_s1, W_s2, b_s2, dlt);

  // 16-17. stage1 min/max normalize + sigmoid mix
  minmax_k<<<dim3(1), dim3(256), 0, stream>>>(s1sc, mmv);
  final_k<<<dim3((KP + 255) / 256), dim3(256), 0, stream>>>(s1sc, dlt, mmv,
                                                            lam, out);
  (void)in_sizes; (void)n_in; (void)out_size; (void)ws_size;
}